// PointPillarsScatter_55637006352904
// MI455X (gfx1250) — compile-verified
//
#include <hip/hip_runtime.h>

typedef float v4f __attribute__((ext_vector_type(4)));
typedef int   v4i __attribute__((ext_vector_type(4)));

// ---------------------------------------------------------------------------
// Kernel 1: init cell->pillar map to -1 (b128 stores, scalar tail)
// ---------------------------------------------------------------------------
__global__ void pp_init_map(int* __restrict__ map, unsigned cells) {
    unsigned t = blockIdx.x * blockDim.x + threadIdx.x;
    unsigned base = t * 4u;
    if (base + 3u < cells) {
        v4i m = { -1, -1, -1, -1 };
        *(v4i*)(map + base) = m;           // map is re-read: keep RT (L2-resident)
    } else if (base < cells) {
        for (unsigned j = base; j < cells; ++j) map[j] = -1;
    }
}

// ---------------------------------------------------------------------------
// Kernel 2: scatter pillar index into map (coords are unique -> race-free)
// ---------------------------------------------------------------------------
__global__ void pp_scatter_map(const int* __restrict__ coords, unsigned P,
                               const int* __restrict__ nxp,
                               const int* __restrict__ nyp,
                               int* __restrict__ map) {
    unsigned p = blockIdx.x * blockDim.x + threadIdx.x;
    if (p >= P) return;
    unsigned nx = (unsigned)nxp[0], ny = (unsigned)nyp[0];
    unsigned b = (unsigned)coords[4u * p + 0u];
    unsigned y = (unsigned)coords[4u * p + 2u];
    unsigned x = (unsigned)coords[4u * p + 3u];
    unsigned cell = b * (ny * nx) + y * nx + x;
    map[cell] = (int)p;
}

// ---------------------------------------------------------------------------
// Kernel 3: output-driven fill. One thread owns 4 consecutive cells.
// Per channel c: 4 b128 feature loads -> 4x4 register transpose ->
// 4 non-temporal b128 stores (coalesced across lanes). One 32-bit udiv
// per thread; same-batch test is a compare (rem0+3 < plane), no 2nd div.
// ---------------------------------------------------------------------------
__global__ void pp_fill(const float* __restrict__ feat,
                        const int* __restrict__ map,
                        float* __restrict__ out,
                        unsigned C, unsigned cells,
                        const int* __restrict__ nxp,
                        const int* __restrict__ nyp) {
    unsigned t = blockIdx.x * blockDim.x + threadIdx.x;
    unsigned cell0 = t * 4u;
    if (cell0 >= cells) return;

    const unsigned plane  = (unsigned)nxp[0] * (unsigned)nyp[0];
    const unsigned Cplane = C * plane;              // fits: <= out_size < 2^31

    const unsigned b0   = cell0 / plane;            // single 32-bit udiv
    const unsigned rem0 = cell0 - b0 * plane;

    const bool full4  = (cell0 + 3u < cells);
    const bool same_b = (rem0 + 3u < plane);

    if (full4 && same_b && ((rem0 & 3u) == 0u) && ((C & 3u) == 0u)) {
        v4i idx = *(const v4i*)(map + cell0);
        // Clamp row pointers so loads are always valid; select value afterwards.
        const float* r0 = feat + (size_t)(unsigned)(idx.x >= 0 ? idx.x : 0) * C;
        const float* r1 = feat + (size_t)(unsigned)(idx.y >= 0 ? idx.y : 0) * C;
        const float* r2 = feat + (size_t)(unsigned)(idx.z >= 0 ? idx.z : 0) * C;
        const float* r3 = feat + (size_t)(unsigned)(idx.w >= 0 ? idx.w : 0) * C;
        // gfx1250 global_prefetch_b8: hide scattered-row latency behind the c-loop
        __builtin_prefetch(r0, 0, 0);
        __builtin_prefetch(r1, 0, 0);
        __builtin_prefetch(r2, 0, 0);
        __builtin_prefetch(r3, 0, 0);

        float* ob = out + (size_t)(b0 * Cplane + rem0);
        const v4f z4 = {0.f, 0.f, 0.f, 0.f};
        for (unsigned c = 0; c < C; c += 4u) {
            v4f a = *(const v4f*)(r0 + c); if (idx.x < 0) a = z4;
            v4f b = *(const v4f*)(r1 + c); if (idx.y < 0) b = z4;
            v4f d = *(const v4f*)(r2 + c); if (idx.z < 0) d = z4;
            v4f e = *(const v4f*)(r3 + c); if (idx.w < 0) e = z4;
            v4f o0 = { a.x, b.x, d.x, e.x };
            v4f o1 = { a.y, b.y, d.y, e.y };
            v4f o2 = { a.z, b.z, d.z, e.z };
            v4f o3 = { a.w, b.w, d.w, e.w };
            float* p0 = ob + (size_t)(c * plane);
            __builtin_nontemporal_store(o0, (v4f*)(p0));
            __builtin_nontemporal_store(o1, (v4f*)(p0 + plane));
            __builtin_nontemporal_store(o2, (v4f*)(p0 + 2u * plane));
            __builtin_nontemporal_store(o3, (v4f*)(p0 + 3u * plane));
        }
    } else {
        // Tail / batch-boundary / misaligned path (rare): incremental b/rem,
        // no further division.
        unsigned b = b0, rem = rem0;
        for (unsigned j = 0; j < 4u; ++j) {
            unsigned cell = cell0 + j;
            if (cell >= cells) break;
            if (rem >= plane) { b += 1u; rem -= plane; }
            int idx = map[cell];
            const float* row = feat + (size_t)(unsigned)(idx >= 0 ? idx : 0) * C;
            float* ob = out + (size_t)(b * Cplane + rem);
            for (unsigned c = 0; c < C; ++c) {
                float v = row[c];
                if (idx < 0) v = 0.f;
                __builtin_nontemporal_store(v, ob + (size_t)(c * plane));
            }
            rem += 1u;
        }
    }
}

// ---------------------------------------------------------------------------
// Fallback path (workspace too small): zero-fill + direct scatter
// ---------------------------------------------------------------------------
__global__ void pp_zero(float* __restrict__ out, unsigned n) {
    unsigned t = blockIdx.x * blockDim.x + threadIdx.x;
    unsigned base = t * 4u;
    if (base + 3u < n) {
        v4f z = {0.f, 0.f, 0.f, 0.f};
        __builtin_nontemporal_store(z, (v4f*)(out + base));
    } else if (base < n) {
        for (unsigned j = base; j < n; ++j)
            __builtin_nontemporal_store(0.f, out + j);
    }
}

__global__ void pp_scatter_direct(const float* __restrict__ feat,
                                  const int* __restrict__ coords,
                                  float* __restrict__ out,
                                  unsigned P, unsigned C,
                                  const int* __restrict__ nxp,
                                  const int* __restrict__ nyp) {
    unsigned p = blockIdx.x * blockDim.x + threadIdx.x;
    if (p >= P) return;
    unsigned nx = (unsigned)nxp[0], ny = (unsigned)nyp[0];
    unsigned plane = nx * ny;
    unsigned b = (unsigned)coords[4u * p + 0u];
    unsigned y = (unsigned)coords[4u * p + 2u];
    unsigned x = (unsigned)coords[4u * p + 3u];
    float* ob = out + (size_t)(b * (C * plane) + y * nx + x);
    const float* row = feat + (size_t)p * C;
    __builtin_prefetch(row, 0, 0);
    for (unsigned c = 0; c < C; ++c)
        ob[(size_t)(c * plane)] = row[c];   // RT store: lines likely L2-resident
}

// ---------------------------------------------------------------------------
extern "C" void kernel_launch(void* const* d_in, const int* in_sizes, int n_in,
                              void* d_out, int out_size, void* d_ws, size_t ws_size,
                              hipStream_t stream) {
    const float* feat   = (const float*)d_in[0];
    const int*   coords = (const int*)d_in[1];
    // d_in[2] = batch_size (derivable; unused), d_in[3] = nx, d_in[4] = ny
    const int* nxp = (const int*)d_in[3];
    const int* nyp = (const int*)d_in[4];

    const int Pi = in_sizes[1] / 4;          // coords is [P,4]
    if (Pi <= 0) return;
    const int Ci = in_sizes[0] / Pi;         // features [P,C]
    if (Ci <= 0) return;
    const unsigned P = (unsigned)Pi;
    const unsigned C = (unsigned)Ci;
    const unsigned cells = (unsigned)out_size / C;   // B*ny*nx
    float* out = (float*)d_out;

    const int BLK = 256;                     // 8 wave32 waves per workgroup

    const bool use_map = ((size_t)cells * sizeof(int)) <= ws_size;

    if (use_map) {
        int* map = (int*)d_ws;
        unsigned q = (cells + 3u) / 4u;
        dim3 gInit((q + BLK - 1) / BLK);
        pp_init_map<<<gInit, BLK, 0, stream>>>(map, cells);

        dim3 gScat((P + BLK - 1) / BLK);
        pp_scatter_map<<<gScat, BLK, 0, stream>>>(coords, P, nxp, nyp, map);

        dim3 gFill((q + BLK - 1) / BLK);
        pp_fill<<<gFill, BLK, 0, stream>>>(feat, map, out, C, cells, nxp, nyp);
    } else {
        unsigned n = (unsigned)out_size;
        unsigned q = (n + 3u) / 4u;
        dim3 gZero((q + BLK - 1) / BLK);
        pp_zero<<<gZero, BLK, 0, stream>>>(out, n);

        dim3 gScat((P + BLK - 1) / BLK);
        pp_scatter_direct<<<gScat, BLK, 0, stream>>>(feat, coords, out, P, C, nxp, nyp);
    }
}